// BasicBlock_14336600834592
// MI455X (gfx1250) — compile-verified
//
#include <hip/hip_runtime.h>
#include <hip/hip_bf16.h>

typedef float v2f __attribute__((ext_vector_type(2)));
typedef float v8f __attribute__((ext_vector_type(8)));

#define NEG_SLOPE 0.01f

// ---------------------------------------------------------------------------
// Zero a float buffer (float4 stores -> global_store_b128)
// ---------------------------------------------------------------------------
__global__ void zero_f4_kernel(float4* __restrict__ p, long long n4) {
  long long i = (long long)blockIdx.x * blockDim.x + threadIdx.x;
  if (i < n4) p[i] = make_float4(0.f, 0.f, 0.f, 0.f);
}

// ---------------------------------------------------------------------------
// Edge scatter-add: agg[dst[e]] += h[src[e]]  (16 lanes per edge, float4 each)
// Gather is global_load_b128; adds are global_atomic_add_f32 (no return),
// which resolve in L2 (agg is 25.6 MB, fully L2-resident on 192 MB L2).
// ---------------------------------------------------------------------------
__global__ void scatter_add_kernel(const float* __restrict__ h,
                                   const int* __restrict__ ei,   // [2, E] int32
                                   float* __restrict__ agg,
                                   int E) {
  long long t = (long long)blockIdx.x * blockDim.x + threadIdx.x;
  int e = (int)(t >> 4);
  if (e >= E) return;
  int f = ((int)t & 15) << 2;                 // feature group: 0,4,...,60
  int s = ei[e];                              // src
  int d = ei[E + e];                          // dst
  float4 v = *(const float4*)(h + (long long)s * 64 + f);
  float* p = agg + (long long)d * 64 + f;
  atomicAdd(p + 0, v.x);
  atomicAdd(p + 1, v.y);
  atomicAdd(p + 2, v.z);
  atomicAdd(p + 3, v.w);
}

// ---------------------------------------------------------------------------
// Fused dual-GEMM + bias + optional residual + LeakyReLU, via
// V_WMMA_F32_16X16X4_F32 (full fp32 precision; GEMM is compute-trivial here).
//
//   out = leaky( A @ Wrel^T + b + H @ Wroot^T (+ resid) )
//
// Block = 256 threads = 8 wave32s. Block covers 32 rows; wave (mt,nt) owns the
// 16x16 tile at rows m0+16*mt, cols 16*nt. K=64 -> 16 k-blocks of 4.
//
// f32 WMMA fragment layouts (ISA 7.12.2):
//   A 16x4 : lanes 0-15 -> M=lane, {V0,V1}={K0,K1}; lanes 16-31 -> {K2,K3}
//   B 4x16 : lanes 0-15 -> N=lane, {V0,V1}={K0,K1}; lanes 16-31 -> {K2,K3}
//   C 16x16: lanes 0-15 -> N=lane, Vr -> M=r; lanes 16-31 -> M=8+r
// B[k][n] = W[n][k], so B fragments index W exactly like A indexes its rows.
// ---------------------------------------------------------------------------
__global__ void gemm_fused_kernel(const float* __restrict__ A,      // [N,64] aggregated
                                  const float* __restrict__ H,      // [N,64] root input
                                  const float* __restrict__ Wrel,   // [64,64]
                                  const float* __restrict__ Wroot,  // [64,64]
                                  const float* __restrict__ bias,   // [64]
                                  const float* __restrict__ resid,  // [N,64] or null
                                  float* __restrict__ out,          // [N,64]
                                  int N) {
  const int wave  = threadIdx.x >> 5;       // 0..7
  const int lane  = threadIdx.x & 31;
  const int mtile = wave >> 2;              // 0..1
  const int ntile = wave & 3;               // 0..3
  const int m0    = blockIdx.x * 32 + mtile * 16;
  const int n0    = ntile * 16;
  const int lrow  = lane & 15;
  const int khalf = (lane >> 4) * 2;        // 0 or 2

  // Clamp row reads so EXEC stays all-1s through the WMMA region.
  int arow = m0 + lrow;
  if (arow > N - 1) arow = N - 1;

  const float* Ap = A     + (long long)arow * 64 + khalf;
  const float* Hp = H     + (long long)arow * 64 + khalf;
  const float* Bp = Wrel  + (long long)(n0 + lrow) * 64 + khalf;
  const float* Cp = Wroot + (long long)(n0 + lrow) * 64 + khalf;

  v8f acc = {};
#pragma unroll
  for (int kb = 0; kb < 16; ++kb) {
    const int k = kb * 4;
    v2f af = *(const v2f*)(Ap + k);
    v2f bf = *(const v2f*)(Bp + k);
    acc = __builtin_amdgcn_wmma_f32_16x16x4_f32(false, af, false, bf,
                                                (short)0, acc, false, false);
    v2f hf = *(const v2f*)(Hp + k);
    v2f cf = *(const v2f*)(Cp + k);
    acc = __builtin_amdgcn_wmma_f32_16x16x4_f32(false, hf, false, cf,
                                                (short)0, acc, false, false);
  }

  const float bv   = bias[n0 + lrow];
  const int   col  = n0 + lrow;
  const int   mbase = m0 + ((lane >> 4) * 8);
#pragma unroll
  for (int r = 0; r < 8; ++r) {
    int m = mbase + r;
    if (m < N) {
      float v = acc[r] + bv;
      if (resid) v += resid[(long long)m * 64 + col];
      v = (v > 0.f) ? v : v * NEG_SLOPE;
      out[(long long)m * 64 + col] = v;
    }
  }
}

// ---------------------------------------------------------------------------
// Launch: zero -> scatter(x) -> gemm1(->h1) -> zero -> scatter(h1) -> gemm2(->out)
// ---------------------------------------------------------------------------
extern "C" void kernel_launch(void* const* d_in, const int* in_sizes, int n_in,
                              void* d_out, int out_size, void* d_ws, size_t ws_size,
                              hipStream_t stream) {
  const float* x       = (const float*)d_in[0];
  const int*   ei      = (const int*)  d_in[1];
  const float* W1_rel  = (const float*)d_in[2];
  const float* b1      = (const float*)d_in[3];
  const float* W1_root = (const float*)d_in[4];
  const float* W2_rel  = (const float*)d_in[5];
  const float* b2      = (const float*)d_in[6];
  const float* W2_root = (const float*)d_in[7];

  const int N = in_sizes[0] / 64;
  const int E = in_sizes[1] / 2;

  float* agg = (float*)d_ws;                      // [N,64]
  float* h1  = agg + (size_t)N * 64;              // [N,64]
  float* out = (float*)d_out;

  const long long n4   = (long long)N * 16;       // float4 count of [N,64]
  const int zero_blks  = (int)((n4 + 255) / 256);
  const int scat_blks  = (int)(((long long)E * 16 + 255) / 256);
  const int gemm_blks  = (N + 31) / 32;

  // Layer 1
  zero_f4_kernel<<<zero_blks, 256, 0, stream>>>((float4*)agg, n4);
  scatter_add_kernel<<<scat_blks, 256, 0, stream>>>(x, ei, agg, E);
  gemm_fused_kernel<<<gemm_blks, 256, 0, stream>>>(agg, x, W1_rel, W1_root, b1,
                                                   nullptr, h1, N);
  // Layer 2 (+ residual x)
  zero_f4_kernel<<<zero_blks, 256, 0, stream>>>((float4*)agg, n4);
  scatter_add_kernel<<<scat_blks, 256, 0, stream>>>(h1, ei, agg, E);
  gemm_fused_kernel<<<gemm_blks, 256, 0, stream>>>(agg, h1, W2_rel, W2_root, b2,
                                                   x, out, N);
}